// MultiScaleDeformableAttnFunction_pytorch_11751030522179
// MI455X (gfx1250) — compile-verified
//
#include <hip/hip_runtime.h>

// MSDeformAttn forward for fixed shapes:
//   value: (16, 8400, 8, 32) f32, levels (80x80, 40x40, 20x20)
//   sampling_locations: (16, 300, 8, 3, 4, 2) f32
//   attention_weights:  (16, 300, 8, 3, 4)    f32
//   out: (16, 300, 256) f32
//
// Pure gather workload: WMMA is structurally inapplicable (per-query B
// matrices, no shared operand). We use the CDNA5 async global->LDS gather
// path (global_load_async_to_lds_b128 + s_wait_asynccnt) to keep all 12
// sample gathers per wave in flight in LDS.

#define NTRIP (16 * 300 * 8)   // 38400 (b,q,h) triples, one wave each
#define WAVES_PER_BLOCK 8
#define NPOINT 12              // L*P
#define ROWF 128               // floats per point buffer (4 corners x 32 ch)

typedef int v4i_ty __attribute__((ext_vector_type(4)));
typedef __attribute__((address_space(1))) v4i_ty gas_v4i;
typedef __attribute__((address_space(3))) v4i_ty las_v4i;

__device__ __forceinline__ void async_gather_b128(const float* gptr,
                                                  float* lds_generic,
                                                  unsigned lds_byte_off) {
#if __has_builtin(__builtin_amdgcn_global_load_async_to_lds_b128)
  __builtin_amdgcn_global_load_async_to_lds_b128(
      (gas_v4i*)gptr, (las_v4i*)lds_generic, /*offset=*/0, /*cpol=*/0);
  (void)lds_byte_off;
#else
  unsigned long long ga = (unsigned long long)gptr;
  asm volatile("global_load_async_to_lds_b128 %0, %1, off"
               :: "v"(lds_byte_off), "v"(ga) : "memory");
  (void)lds_generic;
#endif
}

template <int N>
__device__ __forceinline__ void wait_async() {
#if __has_builtin(__builtin_amdgcn_s_wait_asynccnt)
  __builtin_amdgcn_s_wait_asynccnt((unsigned short)N);
#else
  asm volatile("s_wait_asynccnt %0" :: "i"(N) : "memory");
#endif
  asm volatile("" ::: "memory");  // keep LDS reads below the wait
}

// Compile-time-unrolled consume loop so each wait count is a literal.
template <int J>
__device__ __forceinline__ void consume_points(float4& acc,
                                               const float* ldsWave,
                                               const float (&cw)[NPOINT],
                                               int lane) {
  if constexpr (J < NPOINT) {
    wait_async<NPOINT - 1 - J>();  // async loads complete in order
    const float4 v = *(const float4*)&ldsWave[J * ROWF + lane * 4];
    acc.x += cw[J] * v.x;
    acc.y += cw[J] * v.y;
    acc.z += cw[J] * v.z;
    acc.w += cw[J] * v.w;
    consume_points<J + 1>(acc, ldsWave, cw, lane);
  }
}

__global__ __launch_bounds__(256)
void msda_fwd_kernel(const float* __restrict__ value,
                     const float* __restrict__ sloc,
                     const float* __restrict__ attw,
                     float* __restrict__ out) {
  __shared__ float ldsbuf[WAVES_PER_BLOCK * NPOINT * ROWF];  // 48 KB

  const int lane = threadIdx.x & 31;
  const int wib  = threadIdx.x >> 5;
  const int trip = blockIdx.x * WAVES_PER_BLOCK + wib;  // == (b*300+q)*8+h
  const int b = trip / (300 * 8);
  const int h = trip & 7;

  // lane -> bilinear corner (dx,dy) and 4-channel group
  const int corner = lane >> 3;
  const int dx = corner & 1;
  const int dy = corner >> 1;
  const int cg = lane & 7;

  // Per-triple uniform data: 24 loc floats + 12 weights (16B-aligned).
  const float* locp = sloc + (size_t)trip * 24;
  const float* awp  = attw + (size_t)trip * 12;
  float loc[24];
#pragma unroll
  for (int i = 0; i < 6; ++i) {
    float4 t = ((const float4*)locp)[i];
    loc[4*i+0] = t.x; loc[4*i+1] = t.y; loc[4*i+2] = t.z; loc[4*i+3] = t.w;
  }
  float wgt[12];
#pragma unroll
  for (int i = 0; i < 3; ++i) {
    float4 t = ((const float4*)awp)[i];
    wgt[4*i+0] = t.x; wgt[4*i+1] = t.y; wgt[4*i+2] = t.z; wgt[4*i+3] = t.w;
  }

  // value[b][row][h][c]: row stride 256 floats
  const float* vbase = value + (size_t)b * (8400ull * 256) + h * 32 + cg * 4;
  float* ldsWave = &ldsbuf[wib * (NPOINT * ROWF)];
  const unsigned ldsWaveByte = (unsigned)(wib * (NPOINT * ROWF) * 4);

  const int WW[3]  = {80, 40, 20};
  const int HH[3]  = {80, 40, 20};
  const int OFF[3] = {0, 6400, 8000};

  // Pass 1: compute fused weights + addresses, launch all 12 async gathers.
  float cw[NPOINT];
#pragma unroll
  for (int l = 0; l < 3; ++l) {
    const int W = WW[l], H = HH[l], O = OFF[l];
#pragma unroll
    for (int p = 0; p < 4; ++p) {
      const int j = l * 4 + p;
      const float xf = loc[2*j+0] * (float)W - 0.5f;
      const float yf = loc[2*j+1] * (float)H - 0.5f;
      const float x0f = floorf(xf);
      const float y0f = floorf(yf);
      const float fx = xf - x0f;
      const float fy = yf - y0f;
      const int xi = (int)x0f + dx;
      const int yi = (int)y0f + dy;
      const bool valid = (xi >= 0) & (xi < W) & (yi >= 0) & (yi < H);
      const float bw = (dx ? fx : 1.0f - fx) * (dy ? fy : 1.0f - fy);
      cw[j] = valid ? bw * wgt[j] : 0.0f;
      const int xc = min(max(xi, 0), W - 1);
      const int yc = min(max(yi, 0), H - 1);
      const int row = O + yc * W + xc;
      const int lo = j * ROWF + lane * 4;
      async_gather_b128(vbase + (size_t)row * 256, &ldsWave[lo],
                        ldsWaveByte + (unsigned)lo * 4u);
    }
  }

  // Pass 2: consume in order.
  float4 acc = {0.f, 0.f, 0.f, 0.f};
  consume_points<0>(acc, ldsWave, cw, lane);

  // Reduce the 4 corner partials across lane groups (xor 8, then xor 16).
  acc.x += __shfl_xor(acc.x, 8);
  acc.y += __shfl_xor(acc.y, 8);
  acc.z += __shfl_xor(acc.z, 8);
  acc.w += __shfl_xor(acc.w, 8);
  acc.x += __shfl_xor(acc.x, 16);
  acc.y += __shfl_xor(acc.y, 16);
  acc.z += __shfl_xor(acc.z, 16);
  acc.w += __shfl_xor(acc.w, 16);

  if (lane < 8) {
    *(float4*)(out + (size_t)trip * 32 + cg * 4) = acc;
  }
}

extern "C" void kernel_launch(void* const* d_in, const int* in_sizes, int n_in,
                              void* d_out, int out_size, void* d_ws, size_t ws_size,
                              hipStream_t stream) {
  (void)in_sizes; (void)n_in; (void)out_size; (void)d_ws; (void)ws_size;
  const float* value = (const float*)d_in[0];
  // d_in[1] = value_spatial_shapes (int64) -- compile-time constants here.
  const float* sloc = (const float*)d_in[2];
  const float* attw = (const float*)d_in[3];
  float* out = (float*)d_out;

  dim3 grid(NTRIP / WAVES_PER_BLOCK);  // 4800
  dim3 block(WAVES_PER_BLOCK * 32);    // 256
  hipLaunchKernelGGL(msda_fwd_kernel, grid, block, 0, stream,
                     value, sloc, attw, out);
}